// ConditionalAttentionFusionv2_37331855737580
// MI455X (gfx1250) — compile-verified
//
#include <hip/hip_runtime.h>

// ConditionalAttentionFusion on gfx1250 (MI455X).
// Memory-bound streaming kernel; the per-channel 3x3x2 conv is offloaded to
// V_WMMA_F32_16X16X4_F32 so VALU does not become the bottleneck at the
// 23.3 TB/s-driven element rate. Round 2: boundary handling converted from
// divergent guarded loads (s_and_saveexec chains) to clamp-address +
// v_cndmask selects, so the B/A builds are branch-free.

typedef __attribute__((ext_vector_type(2))) float v2f;
typedef __attribute__((ext_vector_type(8))) float v8f;

#define Bn 4
#define Cn 19
#define Hn 512
#define Wn 1024
#define HWn (Hn * Wn)
#define TILES_X (Wn / 16)              // 64 tiles of 16 pixels along W
#define WAVES_PER_BLOCK 8
#define NUM_TILES (Bn * Hn * TILES_X)  // 131072 wave-tiles

__device__ __forceinline__ int iclamp(int v, int lo, int hi) {
    return v < lo ? lo : (v > hi ? hi : v);
}

__global__ __launch_bounds__(256)
void fusion_wmma_kernel(const float* __restrict__ rgb,
                        const float* __restrict__ dep,
                        const float* __restrict__ rgb_var,
                        const float* __restrict__ d_var,
                        const float* __restrict__ Wp,   // [C,2]   prob fusion
                        const float* __restrict__ Wu,   // [C,2,3,3] -> flat [C][18]
                        const float* __restrict__ Wt,   // [C,2]   total fusion
                        float* __restrict__ out)
{
    const int lane = threadIdx.x & 31;
    const int wave = threadIdx.x >> 5;
    const int tile = blockIdx.x * WAVES_PER_BLOCK + wave;  // grid sized exactly

    // tile -> (b, h, w0)
    const int b   = tile / (Hn * TILES_X);
    const int rem = tile % (Hn * TILES_X);
    const int h   = rem / TILES_X;
    const int w0  = (rem % TILES_X) * 16;

    const int n  = lane & 15;   // pixel within tile (N index)
    const int hi = lane >> 4;   // lane half (selects K/M offsets)

    // Base for streaming rgb/d/out: channel stride = HWn.
    const int baseBH = b * (Cn * HWn) + h * Wn + w0 + n;

    // Prefetch the streaming rgb/d lines for this tile (global_prefetch_b8);
    // overlaps with the cached var-neighborhood gather + WMMA below.
#pragma unroll
    for (int r = 0; r < 8; ++r) {
        const int idx = baseBH + (r + 8 * hi) * HWn;
        __builtin_prefetch(rgb + idx, 0, 3);
        __builtin_prefetch(dep + idx, 0, 3);
    }

    // ---------------- B matrix: neighborhood taps [K=4 x N=16] per K-step ---
    // tap t in [0,18): i = t/9 (0=rgb_var, 1=d_var), kh = (t%9)/3, kw = t%3.
    // Zero for t in {18,19} and SAME-padding borders — implemented branch-free:
    // unconditional load from clamped in-bounds address, then select-to-zero.
    // Layout (16x16x4 f32 B, 4x16): VGPR j, half hi -> K = j + 2*hi, N = lane%16.
    const float* __restrict__ rv = rgb_var + b * HWn;
    const float* __restrict__ dv = d_var  + b * HWn;

    v2f Bm[5];
#pragma unroll
    for (int ks = 0; ks < 5; ++ks) {
#pragma unroll
        for (int j = 0; j < 2; ++j) {
            const int t  = ks * 4 + j + 2 * hi;   // global tap index
            const int r9 = t % 9;                 // harmless garbage for t>=18
            const int hh = h + (r9 / 3) - 1;
            const int ww = w0 + n + (r9 % 3) - 1;
            const bool valid = (t < 18) & (hh >= 0) & (hh < Hn) &
                               (ww >= 0) & (ww < Wn);
            const int hc = iclamp(hh, 0, Hn - 1);
            const int wc = iclamp(ww, 0, Wn - 1);
            const float* __restrict__ vp = (t < 9) ? rv : dv;
            const float v = vp[hc * Wn + wc];     // always in-bounds
            Bm[ks][j] = valid ? v : 0.0f;
        }
    }

    // ---------------- A matrices: conv weights [M=16ch x K=4taps] per K-step -
    // Layout (16x4 f32 A): lane carries M = lane%16; VGPR j, half hi -> K = j + 2*hi.
    // Group g=0 -> channels 0..15, g=1 -> channels 16..18 (rest zero-padded).
    v2f Am[2][5];
#pragma unroll
    for (int g = 0; g < 2; ++g) {
        const int c  = (lane & 15) + 16 * g;
        const int cc = iclamp(c, 0, Cn - 1);
#pragma unroll
        for (int ks = 0; ks < 5; ++ks) {
#pragma unroll
            for (int j = 0; j < 2; ++j) {
                const int t  = ks * 4 + j + 2 * hi;
                const int tt = t > 17 ? 17 : t;
                const float v = Wu[cc * 18 + tt];  // always in-bounds
                Am[g][ks][j] = ((c < Cn) & (t < 18)) ? v : 0.0f;
            }
        }
    }

    // ---------------- WMMA: CDc[c, pix] = sum_t Wu[c,t] * N[t, pix] ---------
    // 5 chained K=4 steps per channel group; EXEC is all-1s here (wave-uniform
    // control flow, exact grid).
    v8f acc0 = {};
    v8f acc1 = {};
#pragma unroll
    for (int ks = 0; ks < 5; ++ks) {
        acc0 = __builtin_amdgcn_wmma_f32_16x16x4_f32(
            false, Am[0][ks], false, Bm[ks], (short)0, acc0, false, false);
        acc1 = __builtin_amdgcn_wmma_f32_16x16x4_f32(
            false, Am[1][ks], false, Bm[ks], (short)0, acc1, false, false);
    }

    // ---------------- Fused elementwise epilogue (streaming phase) ----------
    // D layout: acc[r] -> channel c = r + 8*hi (+16 for group 1), pixel n.
    // out = d + G * (rgb - d),  G = wt0*(wp0*rgb + wp1*d) + wt1*CDc.
#pragma unroll
    for (int r = 0; r < 8; ++r) {                 // group 0: channels 0..15
        const int c   = r + 8 * hi;
        const int idx = baseBH + c * HWn;
        const float x  = rgb[idx];
        const float y  = dep[idx];
        const float ab = Wp[c * 2] * x + Wp[c * 2 + 1] * y;
        const float G  = Wt[c * 2] * ab + Wt[c * 2 + 1] * acc0[r];
        out[idx] = y + G * (x - y);
    }
#pragma unroll
    for (int r = 0; r < 3; ++r) {                 // group 1: channels 16..18
        const int c = 16 + r + 8 * hi;
        if (c < Cn) {                             // only low lane half is valid
            const int idx = baseBH + c * HWn;
            const float x  = rgb[idx];
            const float y  = dep[idx];
            const float ab = Wp[c * 2] * x + Wp[c * 2 + 1] * y;
            const float G  = Wt[c * 2] * ab + Wt[c * 2 + 1] * acc1[r];
            out[idx] = y + G * (x - y);
        }
    }
}

extern "C" void kernel_launch(void* const* d_in, const int* in_sizes, int n_in,
                              void* d_out, int out_size, void* d_ws, size_t ws_size,
                              hipStream_t stream) {
    (void)in_sizes; (void)n_in; (void)out_size; (void)d_ws; (void)ws_size;
    const float* rgb = (const float*)d_in[0];
    const float* dep = (const float*)d_in[1];
    const float* rv  = (const float*)d_in[2];
    const float* dv  = (const float*)d_in[3];
    const float* Wp  = (const float*)d_in[4];
    const float* Wu  = (const float*)d_in[5];
    const float* Wt  = (const float*)d_in[6];
    float* out = (float*)d_out;

    const dim3 block(256, 1, 1);                        // 8 waves (wave32)
    const dim3 grid(NUM_TILES / WAVES_PER_BLOCK, 1, 1); // exact: 16384 blocks
    hipLaunchKernelGGL(fusion_wmma_kernel, grid, block, 0, stream,
                       rgb, dep, rv, dv, Wp, Wu, Wt, out);
}